// GCNNet1_83073257439657
// MI455X (gfx1250) — compile-verified
//
#include <hip/hip_runtime.h>
#include <hip/hip_bf16.h>
#include <stdint.h>

// ---------------------------------------------------------------------------
// GCN (2-layer, DGL-style) for MI455X / gfx1250.
// fp32 WMMA (v_wmma_f32_16x16x4_f32) for all GEMMs; double-buffered
// GLOBAL_LOAD_ASYNC_TO_LDS_B128 weight staging (ASYNCcnt); atomics for edge
// scatter / batchnorm stats / graph pooling.
// ---------------------------------------------------------------------------

#define kN    100000
#define kE    500000
#define kG    1024
#define kIN   128
#define kHID  146
#define kHP   160          // HID padded to multiple of 16 (10 WMMA col tiles)
#define kEPS  1e-5f
#define KC    32           // K-chunk staged in LDS per GEMM block
#define RPB   256          // rows per block in the column-stats kernel

typedef __attribute__((ext_vector_type(2))) float v2f;
typedef __attribute__((ext_vector_type(8))) float v8f;

// ------------------------------- degrees / counts ---------------------------
__global__ void k_degrees(const int* __restrict__ src, const int* __restrict__ dst,
                          float* __restrict__ deg_out, float* __restrict__ deg_in) {
  for (int e = blockIdx.x * blockDim.x + threadIdx.x; e < kE; e += gridDim.x * blockDim.x) {
    atomicAdd(&deg_out[src[e]], 1.0f);
    atomicAdd(&deg_in[dst[e]], 1.0f);
  }
}

__global__ void k_graph_counts(const int* __restrict__ gid, float* __restrict__ counts) {
  for (int i = blockIdx.x * blockDim.x + threadIdx.x; i < kN; i += gridDim.x * blockDim.x)
    atomicAdd(&counts[gid[i]], 1.0f);
}

__global__ void k_rsqrt_clamp(float* __restrict__ d, int n) {
  for (int i = blockIdx.x * blockDim.x + threadIdx.x; i < n; i += gridDim.x * blockDim.x)
    d[i] = rsqrtf(fmaxf(d[i], 1.0f));
}

// --------------------------- weight zero-padding ----------------------------
// Wp[Kpad x kHP] = zero-padded W[Kt x kHID]; makes GEMM weight staging a pure
// contiguous async copy (no guards, 16B-vector friendly).
__global__ void k_pad_weight(const float* __restrict__ W, int Kt, int Kpad,
                             float* __restrict__ Wp) {
  const int total = Kpad * kHP;
  for (int i = blockIdx.x * blockDim.x + threadIdx.x; i < total; i += gridDim.x * blockDim.x) {
    const int k = i / kHP, c = i % kHP;
    Wp[i] = (k < Kt && c < kHID) ? W[(size_t)k * kHID + c] : 0.0f;
  }
}

// ------------------------------- WMMA GEMM ----------------------------------
// C[nrows x kHP] = (rowscale ? diag(rowscale) : I) * A[nrows x apw] @ Bp[Kpad x kHP] (+ bias)
// Bp is pre-padded. Pad columns (>= kHID) of C are written as exactly 0.
// Block: 320 threads = 10 waves; wave w computes the 16x16 tile at cols w*16.
// Weight chunks are double-buffered in LDS via global_load_async_to_lds_b128.
__global__ __launch_bounds__(320)
void k_gemm_wmma(const float* __restrict__ A, int apw, int Kpad,
                 const float* __restrict__ Bp,
                 const float* __restrict__ bias,
                 const float* __restrict__ rowscale,
                 float* __restrict__ C, int nrows) {
  __shared__ __align__(16) float Bs[2][KC * kHP];   // 2 x 20KB

  const int lane = threadIdx.x & 31;
  const int wave = threadIdx.x >> 5;
  const int r0   = blockIdx.x * 16;
  const int c0   = wave * 16;
  const int nlo  = lane & 15;          // M (for A) / N (for B) index within tile
  const int kh   = (lane >> 4) << 1;   // lanes 0-15 -> K{0,1}; lanes 16-31 -> K{2,3}

  const int  arow = r0 + nlo;
  const bool rok  = arow < nrows;
  const float rs  = (rowscale && rok) ? rowscale[arow] : 1.0f;
  const float* __restrict__ Arow = A + (size_t)(rok ? arow : 0) * apw;

  const int t = threadIdx.x;
  // per-chunk slice: 5120 floats = 1280 x float4; 320 threads -> 4 b128 each.
  auto issue_chunk = [&](int kc, int buf) {
    const float* gsrc = Bp + (size_t)kc * kHP + (size_t)t * 4;
    // Low 32 bits of a generic LDS pointer == LDS byte offset (ISA aperture rule).
    const uint32_t lds0 = (uint32_t)(uintptr_t)&Bs[buf][t * 4];
#pragma unroll
    for (int u = 0; u < 4; ++u) {
      asm volatile("global_load_async_to_lds_b128 %0, %1, off"
                   :
                   : "v"(lds0 + (uint32_t)(u * 320 * 16)), "v"(gsrc + u * 320 * 4)
                   : "memory");
    }
  };

  v8f acc = {};
  const int nch = Kpad / KC;           // 4 (emb) or 5 (hidden layers)

  issue_chunk(0, 0);
  for (int ch = 0; ch < nch; ++ch) {
    const int kc  = ch * KC;
    const int buf = ch & 1;
    if (ch + 1 < nch) {
      issue_chunk(kc + KC, buf ^ 1);
      asm volatile("s_wait_asynccnt 0x4" ::: "memory");  // chunk ch complete (in-order)
    } else {
      asm volatile("s_wait_asynccnt 0x0" ::: "memory");
    }
    __syncthreads();

#pragma unroll
    for (int kk = 0; kk < KC; kk += 4) {
      const int kg = kc + kk + kh;     // global K of this lane's first element
      const float2 av = *(const float2*)(Arow + kg);   // 8B aligned (kg even)
      v2f a;
      a.x = rok ? av.x * rs : 0.0f;
      a.y = rok ? av.y * rs : 0.0f;
      v2f b;
      b.x = Bs[buf][(kk + kh)     * kHP + c0 + nlo];
      b.y = Bs[buf][(kk + kh + 1) * kHP + c0 + nlo];
      acc = __builtin_amdgcn_wmma_f32_16x16x4_f32(false, a, false, b, (short)0, acc,
                                                  false, false);
    }
    __syncthreads();   // no thread may still read Bs[buf] when it gets refilled
  }

  // store: VGPR r -> row r0 + r + (lane>=16 ? 8 : 0), col c0 + (lane&15)
  const int col   = c0 + nlo;
  const int rbase = r0 + ((lane >> 4) << 3);
  const float bc  = (bias && col < kHID) ? bias[col] : 0.0f;
#pragma unroll
  for (int r = 0; r < 8; ++r) {
    const int row = rbase + r;
    if (row < nrows) {
      float v = acc[r] + bc;
      if (col >= kHID) v = 0.0f;       // keep pad columns exactly zero
      C[(size_t)row * kHP + col] = v;
    }
  }
}

// ------------------------------- edge scatter -------------------------------
// agg[dst[e]] += m[src[e]]  (float4 per work unit, atomic f32 adds)
__global__ void k_edge_scatter(const float* __restrict__ m,
                               const int* __restrict__ src, const int* __restrict__ dst,
                               float* __restrict__ agg) {
  const int UP = kHP / 4;  // 40 float4 groups per row
  const int total = kE * UP;
  for (int t = blockIdx.x * blockDim.x + threadIdx.x; t < total; t += gridDim.x * blockDim.x) {
    const int e  = t / UP;
    const int c4 = (t % UP) * 4;
    const float4 v = *(const float4*)(m + (size_t)src[e] * kHP + c4);
    float* o = agg + (size_t)dst[e] * kHP + c4;
    atomicAdd(o + 0, v.x);
    atomicAdd(o + 1, v.y);
    atomicAdd(o + 2, v.z);
    atomicAdd(o + 3, v.w);
  }
}

// ------------------- scale by deg_in^-1/2, bias, snorm; column stats --------
// tmp = (agg * rdeg_in[row] + b[c]) * snorm[row]; accumulate sum/sumsq per col.
__global__ __launch_bounds__(kHP)
void k_scale_stats(float* __restrict__ agg, const float* __restrict__ rdeg_in,
                   const float* __restrict__ snorm, const float* __restrict__ bias,
                   float* __restrict__ colstats) {
  const int c = threadIdx.x;                  // 0..159
  const int rstart = blockIdx.x * RPB;
  const int rend = (rstart + RPB < kN) ? (rstart + RPB) : kN;
  const float bc = (c < kHID) ? bias[c] : 0.0f;
  float s = 0.0f, s2 = 0.0f;
  for (int r = rstart; r < rend; ++r) {
    float v = agg[(size_t)r * kHP + c];
    v = (v * rdeg_in[r] + bc) * snorm[r];
    if (c >= kHID) v = 0.0f;
    agg[(size_t)r * kHP + c] = v;
    s += v;
    s2 += v * v;
  }
  atomicAdd(&colstats[c], s);
  atomicAdd(&colstats[kHP + c], s2);
}

// ---------------- batchnorm (batch stats) + ReLU + residual -----------------
__global__ void k_norm_relu_res(const float* __restrict__ tmp, const float* __restrict__ h_in,
                                const float* __restrict__ colstats,
                                const float* __restrict__ gamma, const float* __restrict__ beta,
                                float* __restrict__ h_out) {
  const int total = kN * kHP;
  const float invN = 1.0f / (float)kN;
  for (int i = blockIdx.x * blockDim.x + threadIdx.x; i < total; i += gridDim.x * blockDim.x) {
    const int c = i % kHP;
    float o = 0.0f;
    if (c < kHID) {
      const float mu  = colstats[c] * invN;
      const float var = colstats[kHP + c] * invN - mu * mu;
      float v = (tmp[i] - mu) * rsqrtf(var + kEPS) * gamma[c] + beta[c];
      v = fmaxf(v, 0.0f);
      o = h_in[i] + v;
    }
    h_out[i] = o;
  }
}

// ------------------------------- graph mean pool ----------------------------
__global__ void k_pool_accum(const float* __restrict__ h, const int* __restrict__ gid,
                             float* __restrict__ pool) {
  const int total = kN * kHP;
  for (int i = blockIdx.x * blockDim.x + threadIdx.x; i < total; i += gridDim.x * blockDim.x) {
    const int row = i / kHP, c = i % kHP;
    atomicAdd(&pool[(size_t)gid[row] * kHP + c], h[i]);
  }
}

__global__ void k_pool_final(const float* __restrict__ pool, const float* __restrict__ counts,
                             float* __restrict__ out) {
  const int total = kG * kHID;
  for (int i = blockIdx.x * blockDim.x + threadIdx.x; i < total; i += gridDim.x * blockDim.x) {
    const int g = i / kHID, c = i % kHID;
    out[i] = pool[(size_t)g * kHP + c] / fmaxf(counts[g], 1.0f);
  }
}

// ---------------------------------------------------------------------------
extern "C" void kernel_launch(void* const* d_in, const int* in_sizes, int n_in,
                              void* d_out, int out_size, void* d_ws, size_t ws_size,
                              hipStream_t stream) {
  (void)in_sizes; (void)n_in; (void)out_size; (void)ws_size;

  const float* nodes = (const float*)d_in[0];
  const float* snorm = (const float*)d_in[1];
  const float* W_emb = (const float*)d_in[2];
  const float* b_emb = (const float*)d_in[3];
  const float* W1    = (const float*)d_in[4];
  const float* b1    = (const float*)d_in[5];
  const float* g1    = (const float*)d_in[6];
  const float* be1   = (const float*)d_in[7];
  const float* W2    = (const float*)d_in[8];
  const float* b2    = (const float*)d_in[9];
  const float* g2    = (const float*)d_in[10];
  const float* be2   = (const float*)d_in[11];
  const int*   src   = (const int*)d_in[12];
  const int*   dst   = (const int*)d_in[13];
  const int*   gid   = (const int*)d_in[14];

  float* out = (float*)d_out;

  // workspace layout (floats; base is 256B aligned, all offsets keep 16B align)
  float* bufX = (float*)d_ws;                       // h0 -> (layer2 m) -> h2
  float* bufY = bufX + (size_t)kN * kHP;            // layer1 m -> h1
  float* bufZ = bufY + (size_t)kN * kHP;            // agg / tmp
  float* rdo  = bufZ + (size_t)kN * kHP;            // deg_out^-1/2
  float* rdi  = rdo + kN;                           // deg_in^-1/2
  float* cst  = rdi + kN;                           // colstats [2*kHP]
  float* cnts = cst + 2 * kHP;                      // graph counts [kG]
  float* pool = cnts + kG;                          // [kG * kHP]
  float* wpe  = pool + (size_t)kG * kHP;            // padded W_emb [kIN x kHP]
  float* wp1  = wpe + (size_t)kIN * kHP;            // padded W1   [kHP x kHP]
  float* wp2  = wp1 + (size_t)kHP * kHP;            // padded W2   [kHP x kHP]

  const int TB = 256;
  const int gN   = (kN + TB - 1) / TB;              // 391
  const int gE   = (kE + TB - 1) / TB;              // 1954
  const int gNH  = (kN * kHP + TB - 1) / TB;        // 62500
  const int gSc  = (kE * (kHP / 4) + TB - 1) / TB;  // 78125
  const int gStat = (kN + RPB - 1) / RPB;           // 391
  const int gGemm = (kN + 15) / 16;                 // 6250
  const int gPoolF = (kG * kHID + TB - 1) / TB;
  const int gPadE = (kIN * kHP + TB - 1) / TB;
  const int gPadH = (kHP * kHP + TB - 1) / TB;

  // degrees + counts + pool init + weight padding
  hipMemsetAsync(rdo, 0, 2 * (size_t)kN * sizeof(float), stream);
  hipMemsetAsync(cnts, 0, (size_t)(kG + kG * kHP) * sizeof(float), stream);
  k_degrees<<<gE, TB, 0, stream>>>(src, dst, rdo, rdi);
  k_graph_counts<<<gN, TB, 0, stream>>>(gid, cnts);
  k_rsqrt_clamp<<<gN, TB, 0, stream>>>(rdo, kN);
  k_rsqrt_clamp<<<gN, TB, 0, stream>>>(rdi, kN);
  k_pad_weight<<<gPadE, TB, 0, stream>>>(W_emb, kIN, kIN, wpe);
  k_pad_weight<<<gPadH, TB, 0, stream>>>(W1, kHID, kHP, wp1);
  k_pad_weight<<<gPadH, TB, 0, stream>>>(W2, kHID, kHP, wp2);

  // embedding: h0 = nodes @ W_emb + b_emb   -> bufX
  k_gemm_wmma<<<gGemm, 320, 0, stream>>>(nodes, kIN, kIN, wpe, b_emb, nullptr,
                                         bufX, kN);

  // ---------------- layer 1 : in=bufX, m=bufY, agg=bufZ, out=bufY ----------
  k_gemm_wmma<<<gGemm, 320, 0, stream>>>(bufX, kHP, kHP, wp1, nullptr, rdo,
                                         bufY, kN);
  hipMemsetAsync(bufZ, 0, (size_t)kN * kHP * sizeof(float), stream);
  k_edge_scatter<<<gSc, TB, 0, stream>>>(bufY, src, dst, bufZ);
  hipMemsetAsync(cst, 0, 2 * kHP * sizeof(float), stream);
  k_scale_stats<<<gStat, kHP, 0, stream>>>(bufZ, rdi, snorm, b1, cst);
  k_norm_relu_res<<<gNH, TB, 0, stream>>>(bufZ, bufX, cst, g1, be1, bufY);

  // ---------------- layer 2 : in=bufY, m=bufX, agg=bufZ, out=bufX ----------
  k_gemm_wmma<<<gGemm, 320, 0, stream>>>(bufY, kHP, kHP, wp2, nullptr, rdo,
                                         bufX, kN);
  hipMemsetAsync(bufZ, 0, (size_t)kN * kHP * sizeof(float), stream);
  k_edge_scatter<<<gSc, TB, 0, stream>>>(bufX, src, dst, bufZ);
  hipMemsetAsync(cst, 0, 2 * kHP * sizeof(float), stream);
  k_scale_stats<<<gStat, kHP, 0, stream>>>(bufZ, rdi, snorm, b2, cst);
  k_norm_relu_res<<<gNH, TB, 0, stream>>>(bufZ, bufY, cst, g2, be2, bufX);

  // ---------------- graph mean pooling -> d_out ----------------------------
  k_pool_accum<<<gNH, TB, 0, stream>>>(bufX, gid, pool);
  k_pool_final<<<gPoolF, TB, 0, stream>>>(pool, cnts, out);
}